// MultiBoxLoss_8340826488891
// MI455X (gfx1250) — compile-verified
//
#include <hip/hip_runtime.h>
#include <cstdint>
#include <cstddef>

// Problem sizes (fixed by the reference setup_inputs)
#define N_B   32
#define P_N   18000
#define C_N   80
#define O_N   16
#define TOPK_ 15
#define BLK   256
#define PBLK  ((P_N + BLK - 1) / BLK)      // 71 blocks along P
#define NPART (N_B * PBLK)                 // 2272 per-(n,pblock) partials
#define QC    (C_N / 4)                    // 20 float4 per (n,p) conf row
#define CLSB  ((N_B * P_N * QC) / BLK)     // 45000, exact

typedef float v2f __attribute__((ext_vector_type(2)));
typedef float v8f __attribute__((ext_vector_type(8)));

#define ASG __attribute__((address_space(1)))
#define ASL __attribute__((address_space(3)))

#if __has_builtin(__builtin_amdgcn_global_load_async_to_lds_b32)
#define HAVE_ASYNC_LDS 1
#endif

// ---- workspace layout (units: floats) ----
#define OFF_REG   ((size_t)0)
#define OFF_PRED  ((size_t)N_B * O_N * P_N)              // 9,216,000
#define OFF_TVAL  ((size_t)2 * N_B * O_N * P_N)          // 18,432,000
#define OFF_TCLS  (OFF_TVAL + (size_t)N_B * P_N)
#define OFF_PSW   (OFF_TCLS + (size_t)N_B * P_N)
#define OFF_PNP   (OFF_PSW + (size_t)NPART)
#define OFF_PLOC  (OFF_PNP + (size_t)NPART)
#define OFF_PCLS  (OFF_PLOC + (size_t)NPART)
#define WS_FLOATS (OFF_PCLS + (size_t)CLSB)

// ---------------------------------------------------------------------------
// Exact f32 wave reduction on the matrix pipe (verified: lowers to
// v_wmma_f32_16x16x4_f32). A(16x4) x ones(4x16): D[m][*] = v_m + v_{m+16};
// in-lane sum of the 8 accumulator VGPRs + lane^16 fold = full 32-lane sum,
// pure f32, deterministic.
// ---------------------------------------------------------------------------
__device__ __forceinline__ float wave_sum_wmma(float v) {
    v2f a = {v, 0.0f};
    v2f b = {1.0f, 1.0f};
    v8f c = {0.f, 0.f, 0.f, 0.f, 0.f, 0.f, 0.f, 0.f};
    v8f d = __builtin_amdgcn_wmma_f32_16x16x4_f32(
        false, a, false, b, (short)0, c, false, false);
    float s = d[0] + d[1] + d[2] + d[3] + d[4] + d[5] + d[6] + d[7];
    s += __shfl_xor(s, 16, 32);
    return s;
}

// Block-wide sum (BLK==256 -> 8 waves). Valid on thread 0. Call from uniform
// control flow only (WMMA requires full EXEC).
__device__ __forceinline__ float block_sum(float v, float* sh) {
    float s = wave_sum_wmma(v);
    int lane = threadIdx.x & 31;
    int wv   = threadIdx.x >> 5;
    if (lane == 0) sh[wv] = s;
    __syncthreads();
    float r = 0.f;
    if (threadIdx.x == 0) {
#pragma unroll
        for (int i = 0; i < BLK / 32; ++i) r += sh[i];
    }
    __syncthreads();   // allow sh reuse by a following block_sum
    return r;
}

// Stage one batch item's truths(64 f32) + labels(16 i32) into LDS, using the
// CDNA5 async global->LDS path (builtin takes b32-typed int* pointers:
// global AS1 source, LDS AS3 destination).
__device__ __forceinline__ void stage_row(const float* __restrict__ gt,
                                          const int* __restrict__ gl,
                                          float* sh_t, int* sh_l) {
    const int t = threadIdx.x;
#ifdef HAVE_ASYNC_LDS
    if (t < O_N * 4)
        __builtin_amdgcn_global_load_async_to_lds_b32(
            (ASG int*)(gt + t), (ASL int*)(sh_t + t), 0, 0);
    if (t < O_N)
        __builtin_amdgcn_global_load_async_to_lds_b32(
            (ASG int*)(gl + t), (ASL int*)(sh_l + t), 0, 0);
#if __has_builtin(__builtin_amdgcn_s_wait_asynccnt)
    __builtin_amdgcn_s_wait_asynccnt(0);
#else
    asm volatile("s_wait_asynccnt 0" ::: "memory");
#endif
#else
    if (t < O_N * 4) sh_t[t] = gt[t];
    if (t < O_N)     sh_l[t] = gl[t];
#endif
    __syncthreads();
}

__device__ __forceinline__ float iou_xyxy(float ax0, float ay0, float ax1, float ay1,
                                          float bx0, float by0, float bx1, float by1) {
    float ix = fmaxf(fminf(ax1, bx1) - fmaxf(ax0, bx0), 0.f);
    float iy = fmaxf(fminf(ay1, by1) - fmaxf(ay0, by0), 0.f);
    float inter = ix * iy;
    float aa = (ax1 - ax0) * (ay1 - ay0);
    float ab = (bx1 - bx0) * (by1 - by0);
    return inter / (aa + ab - inter);
}

__device__ __forceinline__ float balanced_l1(float diff) {
    const float alpha = 0.5f, gamma = 1.5f, beta = 0.11f;
    const float bb = 19.0855369f;       // e^(gamma/alpha) - 1
    float d = fabsf(diff);
    float small = alpha / bb * (bb * d + 1.0f) * log1pf(bb * d / beta) - alpha * d;
    float big   = gamma * d + gamma / bb - alpha * beta;
    return (d < beta) ? small : big;
}

__device__ __forceinline__ float gfocal_elem(float x, float t) {
    float bce  = fmaxf(x, 0.f) - x * t + log1pf(expf(-fabsf(x)));
    float sig  = 1.f / (1.f + expf(-x));
    float diff = sig - t;
    return diff * diff * bce;
}

// ---------------------------------------------------------------------------
// K1: build reg_ov and pred [N,O,P] with column-max masking applied.
// ---------------------------------------------------------------------------
__global__ void k_match(const float* __restrict__ loc,
                        const float* __restrict__ conf,
                        const float* __restrict__ priors,
                        const float* __restrict__ truths,
                        const int*   __restrict__ labels,
                        float* __restrict__ reg,
                        float* __restrict__ pred) {
    const int n = blockIdx.y;
    const int p = blockIdx.x * BLK + threadIdx.x;

    __shared__ float sh_t[O_N * 4];
    __shared__ int   sh_l[O_N];
    stage_row(truths + (size_t)n * O_N * 4, labels + n * O_N, sh_t, sh_l);
    if (p >= P_N) return;

    float4 pr = ((const float4*)priors)[p];                         // cx,cy,w,h
    float4 lo = ((const float4*)loc)[(size_t)n * P_N + p];

    // decode(loc, prior) -> xyxy
    float dcx = pr.x + lo.x * 0.1f * pr.z;
    float dcy = pr.y + lo.y * 0.1f * pr.w;
    float dw  = pr.z * expf(lo.z * 0.2f);
    float dh  = pr.w * expf(lo.w * 0.2f);
    float dx0 = dcx - dw * 0.5f, dy0 = dcy - dh * 0.5f;
    float dx1 = dcx + dw * 0.5f, dy1 = dcy + dh * 0.5f;
    // point_form(prior)
    float fx0 = pr.x - pr.z * 0.5f, fy0 = pr.y - pr.w * 0.5f;
    float fx1 = pr.x + pr.z * 0.5f, fy1 = pr.y + pr.w * 0.5f;

    const float* cbase = conf + ((size_t)n * P_N + p) * C_N;

    float rv[O_N], qv[O_N];
    float rmax = -1.f, qmax = -1.f;
#pragma unroll
    for (int o = 0; o < O_N; ++o) {
        float tx0 = sh_t[o * 4 + 0], ty0 = sh_t[o * 4 + 1];
        float tx1 = sh_t[o * 4 + 2], ty1 = sh_t[o * 4 + 3];
        float r  = iou_xyxy(tx0, ty0, tx1, ty1, dx0, dy0, dx1, dy1);
        float ov = iou_xyxy(tx0, ty0, tx1, ty1, fx0, fy0, fx1, fy1);
        float pc = 1.f / (1.f + expf(-cbase[sh_l[o]]));
        float q  = powf(ov, (2.0f - pc) * 0.5f);   // ov ** ((SIGMA-pc)/SIGMA)
        rv[o] = r; qv[o] = q;
        rmax = fmaxf(rmax, r);
        qmax = fmaxf(qmax, q);
    }
#pragma unroll
    for (int o = 0; o < O_N; ++o) {
        size_t idx = ((size_t)n * O_N + o) * P_N + p;
        reg[idx]  = (rv[o] == rmax) ? rv[o] : 0.f;
        pred[idx] = (qv[o] == qmax) ? qv[o] : 0.f;
    }
}

// ---------------------------------------------------------------------------
// K2: single-pass boost_topk per row (covers both matrices; 2*N*O blocks).
// Each thread keeps a sorted local top-15 of its strided elements in its LDS
// slice (every thread owns >= 70 elements, so all 15 slots are real), then a
// log2(256)-step tournament merge of sorted lists. Ranking: value desc, index
// asc on ties -- identical selection to lax.top_k + iterative argmax.
// ---------------------------------------------------------------------------
__global__ void k_boost(float* __restrict__ reg, float* __restrict__ pred) {
    const int row = blockIdx.x;          // 0 .. 2*N*O-1
    float* mat = (row < N_B * O_N) ? (reg + (size_t)row * P_N)
                                   : (pred + (size_t)(row - N_B * O_N) * P_N);
    __shared__ float lv[BLK * TOPK_];    // 15,360 B
    __shared__ int   li[BLK * TOPK_];    // 15,360 B
    const int tid = threadIdx.x;
    float* mv = lv + tid * TOPK_;
    int*   mi = li + tid * TOPK_;

#pragma unroll
    for (int k = 0; k < TOPK_; ++k) { mv[k] = -1.f; mi[k] = 0x7fffffff; }

    for (int j = tid; j < P_N; j += BLK) {
        __builtin_prefetch(&mat[j + BLK], 0, 0);         // global_prefetch_b8
        float nv = mat[j];
        if (nv > mv[TOPK_ - 1]) {        // equal value -> existing (lower idx) wins
            int pos = TOPK_ - 1;
            while (pos > 0 && nv > mv[pos - 1]) {
                mv[pos] = mv[pos - 1]; mi[pos] = mi[pos - 1]; --pos;
            }
            mv[pos] = nv; mi[pos] = j;
        }
    }
    __syncthreads();

    for (int s = BLK / 2; s >= 1; s >>= 1) {
        if (tid < s) {
            const float* av  = lv + tid * TOPK_;
            const int*   ai  = li + tid * TOPK_;
            const float* bv2 = lv + (tid + s) * TOPK_;
            const int*   bi2 = li + (tid + s) * TOPK_;
            float ov_[TOPK_]; int oi_[TOPK_];
            int ia = 0, ib = 0;
#pragma unroll
            for (int k = 0; k < TOPK_; ++k) {
                float avv = av[ia], bvv = bv2[ib];
                bool ta = (avv > bvv) || (avv == bvv && ai[ia] < bi2[ib]);
                if (ta) { ov_[k] = avv; oi_[k] = ai[ia]; ++ia; }
                else    { ov_[k] = bvv; oi_[k] = bi2[ib]; ++ib; }
            }
#pragma unroll
            for (int k = 0; k < TOPK_; ++k) {
                lv[tid * TOPK_ + k] = ov_[k];
                li[tid * TOPK_ + k] = oi_[k];
            }
        }
        __syncthreads();
    }

    if (tid == 0) {
        float ssum = 0.f;
        for (int k = 0; k < TOPK_; ++k) ssum += lv[k];
        int np = (int)ssum;              // trunc, like .astype(int32)
        if (np < 1) np = 1;
        if (np > TOPK_) np = TOPK_;
        for (int k = 0; k < np; ++k) mat[li[k]] += 3.0f;
    }
}

// ---------------------------------------------------------------------------
// K3: per-(n,p) finalize: overlap_t / conf_t / pred_t / loc_t, encode target,
// loc-loss partial, weight-sum partial, num_pos partial.
// ---------------------------------------------------------------------------
__global__ void k_final(const float* __restrict__ loc,
                        const float* __restrict__ priors,
                        const float* __restrict__ truths,
                        const int*   __restrict__ labels,
                        const float* __restrict__ reg,
                        const float* __restrict__ pred,
                        float* __restrict__ tval,
                        int*   __restrict__ tcls,
                        float* __restrict__ p_sumw,
                        float* __restrict__ p_npos,
                        float* __restrict__ p_loc) {
    const int n = blockIdx.y;
    const int p = blockIdx.x * BLK + threadIdx.x;

    __shared__ float sh_t[O_N * 4];
    __shared__ int   sh_l[O_N];
    __shared__ float sh_red[BLK / 32];
    stage_row(truths + (size_t)n * O_N * 4, labels + n * O_N, sh_t, sh_l);

    float wv = 0.f, npos = 0.f, locl = 0.f;
    if (p < P_N) {
        float ovt = -1.f; int am = 0;
        float pt  = -1.f; int ap = 0;
#pragma unroll
        for (int o = 0; o < O_N; ++o) {
            size_t idx = ((size_t)n * O_N + o) * P_N + p;
            float rv = reg[idx];
            if (rv > ovt) { ovt = rv; am = o; }      // first max, like jnp.argmax
            float qv = pred[idx];
            if (qv > pt)  { pt = qv;  ap = o; }
        }
        tval[(size_t)n * P_N + p] = fmaxf(ovt - 3.f, 0.f);
        tcls[(size_t)n * P_N + p] = (ovt > 1.f) ? sh_l[am] : -1;
        npos = (ovt > 3.f) ? 1.f : 0.f;
        wv   = fmaxf(pt - 3.f, 0.f);

        // encode(loc_t = truths[ap], prior) and balanced L1 vs loc_data
        float4 pr = ((const float4*)priors)[p];
        float tx0 = sh_t[ap * 4 + 0], ty0 = sh_t[ap * 4 + 1];
        float tx1 = sh_t[ap * 4 + 2], ty1 = sh_t[ap * 4 + 3];
        float gcx = ((tx0 + tx1) * 0.5f - pr.x) / (0.1f * pr.z);
        float gcy = ((ty0 + ty1) * 0.5f - pr.y) / (0.1f * pr.w);
        float gw  = logf((tx1 - tx0) / pr.z) / 0.2f;
        float gh  = logf((ty1 - ty0) / pr.w) / 0.2f;
        float4 lo = ((const float4*)loc)[(size_t)n * P_N + p];
        locl = wv * (balanced_l1(lo.x - gcx) + balanced_l1(lo.y - gcy) +
                     balanced_l1(lo.z - gw)  + balanced_l1(lo.w - gh));
    }
    // uniform control flow from here: WMMA reductions
    const int pb = blockIdx.y * gridDim.x + blockIdx.x;
    float s1 = block_sum(wv,   sh_red);
    float s2 = block_sum(npos, sh_red);
    float s3 = block_sum(locl, sh_red);
    if (threadIdx.x == 0) { p_sumw[pb] = s1; p_npos[pb] = s2; p_loc[pb] = s3; }
}

// ---------------------------------------------------------------------------
// K4: gfocal over all of conf_data (float4 per thread, exact grid).
// ---------------------------------------------------------------------------
__global__ void k_cls(const float* __restrict__ conf,
                      const float* __restrict__ tval,
                      const int*   __restrict__ tcls,
                      float* __restrict__ p_cls) {
    __shared__ float sh_red[BLK / 32];
    const int e = blockIdx.x * BLK + threadIdx.x;          // float4 index
    const int n = e / (P_N * QC);
    const int r = e % (P_N * QC);
    const int p = r / QC;
    const int q = r % QC;

    float  tv = tval[(size_t)n * P_N + p];
    int    tc = tcls[(size_t)n * P_N + p];
    float4 cf = ((const float4*)conf)[e];
    const int c0 = q * 4;

    float acc = 0.f;
    acc += gfocal_elem(cf.x, (c0 + 0 == tc) ? tv : 0.f);
    acc += gfocal_elem(cf.y, (c0 + 1 == tc) ? tv : 0.f);
    acc += gfocal_elem(cf.z, (c0 + 2 == tc) ? tv : 0.f);
    acc += gfocal_elem(cf.w, (c0 + 3 == tc) ? tv : 0.f);

    float s = block_sum(acc, sh_red);
    if (threadIdx.x == 0) p_cls[blockIdx.x] = s;
}

// ---------------------------------------------------------------------------
// K5: final fold — deterministic, single block.
// ---------------------------------------------------------------------------
__global__ void k_out(const float* __restrict__ p_sumw,
                      const float* __restrict__ p_npos,
                      const float* __restrict__ p_loc,
                      const float* __restrict__ p_cls,
                      float* __restrict__ out) {
    __shared__ float sh_red[BLK / 32];
    const int tid = threadIdx.x;

    float a = 0.f;
    for (int i = tid; i < NPART; i += BLK) a += p_sumw[i];
    float sumw = block_sum(a, sh_red);

    a = 0.f;
    for (int i = tid; i < NPART; i += BLK) a += p_npos[i];
    float npos = block_sum(a, sh_red);

    a = 0.f;
    for (int i = tid; i < NPART; i += BLK) a += p_loc[i];
    float lossl = block_sum(a, sh_red);

    a = 0.f;
    for (int i = tid; i < CLSB; i += BLK) a += p_cls[i];
    float lossc = block_sum(a, sh_red);

    if (tid == 0) {
        out[0] = lossl / (4.0f * sumw);                  // w normalized by w.sum()
        out[1] = lossc / fmaxf(npos, 1.0f);              // / max(num_pos, 1)
    }
}

// ---------------------------------------------------------------------------
extern "C" void kernel_launch(void* const* d_in, const int* in_sizes, int n_in,
                              void* d_out, int out_size, void* d_ws, size_t ws_size,
                              hipStream_t stream) {
    (void)in_sizes; (void)n_in; (void)out_size;
    if (ws_size < WS_FLOATS * sizeof(float)) return;     // workspace guard

    const float* loc    = (const float*)d_in[0];   // [N,P,4]
    const float* conf   = (const float*)d_in[1];   // [N,P,C]
    const float* priors = (const float*)d_in[2];   // [P,4]
    const float* truths = (const float*)d_in[3];   // [N,O,4]
    const int*   labels = (const int*)d_in[4];     // [N,O]
    float* out = (float*)d_out;

    float* W     = (float*)d_ws;
    float* reg   = W + OFF_REG;
    float* pred  = W + OFF_PRED;
    float* tval  = W + OFF_TVAL;
    int*   tcls  = (int*)(W + OFF_TCLS);
    float* psw   = W + OFF_PSW;
    float* pnp   = W + OFF_PNP;
    float* ploc  = W + OFF_PLOC;
    float* pcls  = W + OFF_PCLS;

    dim3 gNP(PBLK, N_B);
    k_match<<<gNP, BLK, 0, stream>>>(loc, conf, priors, truths, labels, reg, pred);
    k_boost<<<2 * N_B * O_N, BLK, 0, stream>>>(reg, pred);
    k_final<<<gNP, BLK, 0, stream>>>(loc, priors, truths, labels, reg, pred,
                                     tval, tcls, psw, pnp, ploc);
    k_cls<<<CLSB, BLK, 0, stream>>>(conf, tval, tcls, pcls);
    k_out<<<1, BLK, 0, stream>>>(psw, pnp, ploc, pcls, out);
}